// DCNv3_1D_3710851744516
// MI455X (gfx1250) — compile-verified
//
#include <hip/hip_runtime.h>
#include <hip/hip_bf16.h>

typedef __attribute__((ext_vector_type(2))) float v2f;
typedef __attribute__((ext_vector_type(8))) float v8f;

#define C_DIM   256
#define L_DIM   4096
#define N_DIM   16
#define G_DIM   8
#define GC_DIM  32
#define K_TAPS  3
#define M_DIM   (N_DIM * L_DIM)   // 65536 rows for every GEMM

// ---------------------------------------------------------------------------
// Out[M, NOUT] = A[M, KD] @ W[NOUT, KD]^T + bias       (all fp32)
// One wave computes a 16(M) x 64(N) tile via V_WMMA_F32_16X16X4_F32.
// All shape params are template constants so the subtile loop fully unrolls
// into SUBT distinct WMMAs with statically allocated v8f accumulators
// (no movrel indexed-VGPR access), and the NOUT=24 partial-tile guard is a
// hoisted branch-free select.
//
// A-frag layout (ISA 7.12.2, 32-bit A 16x4): lane&15 -> M row, lane>>4
// selects the K-pair, VGPR0/1 = even/odd K -> one global_load_b64 per step.
// B-frag mirrors it (lane&15 -> N col); W is row-major [NOUT,KD] so W rows
// are exactly the B columns (NT GEMM, no transpose needed).
// ---------------------------------------------------------------------------
template <int NOUT, int KD>
__global__ __launch_bounds__(256) void gemm_nt_wmma_f32(
    const float* __restrict__ A, const float* __restrict__ W,
    const float* __restrict__ bias, float* __restrict__ Out)
{
  constexpr int NTILES_N = (NOUT + 63) / 64;               // waves per M-tile
  constexpr int SUBT     = (NOUT >= 64) ? 4 : ((NOUT + 15) / 16);
  constexpr bool PARTIAL = (NOUT % 16) != 0;
  static_assert(NOUT % 64 == 0 || NOUT < 64, "tile scheme assumption");

  const int lane   = threadIdx.x & 31;
  const int wave   = blockIdx.x * (blockDim.x >> 5) + (threadIdx.x >> 5);
  const int mt     = wave / NTILES_N;
  const int nt     = wave - mt * NTILES_N;
  const int m_base = mt * 16;
  if (m_base >= M_DIM) return;          // uniform per wave: EXEC stays full
  const int n0    = nt * 64;
  const int mrow  = lane & 15;
  const int khalf = lane >> 4;
  const v2f vzero = {0.f, 0.f};

  v8f acc[SUBT];
#pragma unroll
  for (int t = 0; t < SUBT; ++t) acc[t] = v8f{};

  const float* aptr = A + (size_t)(m_base + mrow) * KD + 2 * khalf;

  // Hoisted per-subtile B pointers; invalid lanes clamp to row 0 and get
  // their loaded value zeroed by a select (no per-iteration branching).
  const float* wptr[SUBT];
  bool         wlive[SUBT];
#pragma unroll
  for (int t = 0; t < SUBT; ++t) {
    const int nc = n0 + t * 16 + mrow;
    wlive[t] = (nc < NOUT);
    wptr[t]  = W + (size_t)(wlive[t] ? nc : 0) * KD + 2 * khalf;
  }

#pragma unroll 2
  for (int k = 0; k < KD; k += 4) {
    const v2f a = *(const v2f*)(aptr + k);
#pragma unroll
    for (int t = 0; t < SUBT; ++t) {
      v2f b = *(const v2f*)(wptr[t] + k);
      if (PARTIAL && t == SUBT - 1 && !wlive[t]) b = vzero;
      acc[t] = __builtin_amdgcn_wmma_f32_16x16x4_f32(
          false, a, false, b, (short)0, acc[t], false, false);
    }
  }

#pragma unroll
  for (int t = 0; t < SUBT; ++t) {
    const int nc = n0 + t * 16 + mrow;
    if (!PARTIAL || nc < NOUT) {
      const float bv = bias[nc];
#pragma unroll
      for (int r = 0; r < 8; ++r) {     // C/D layout: row = r + 8*khalf
        const int row = m_base + r + khalf * 8;
        Out[(size_t)row * NOUT + nc] = acc[t][r] + bv;
      }
    }
  }
}

// ---------------------------------------------------------------------------
// Depthwise conv1d, K=3, dilation 1, zero 'same' padding:
// feat[n,l,c] = b[c] + sum_k w[c,k] * x[n, l+k-1, c]
// ---------------------------------------------------------------------------
__global__ __launch_bounds__(256) void dwconv3_kernel(
    const float* __restrict__ x, const float* __restrict__ dww,
    const float* __restrict__ dwb, float* __restrict__ feat)
{
  const size_t idx = (size_t)blockIdx.x * blockDim.x + threadIdx.x;
  const int c = (int)(idx & (C_DIM - 1));
  const int l = (int)((idx >> 8) & (L_DIM - 1));
  float acc = dwb[c] + dww[c * 3 + 1] * x[idx];
  if (l > 0)          acc += dww[c * 3 + 0] * x[idx - C_DIM];
  if (l < L_DIM - 1)  acc += dww[c * 3 + 2] * x[idx + C_DIM];
  feat[idx] = acc;
}

// ---------------------------------------------------------------------------
// Deformable sampling: per (n,l,c): loc = l + (k-1) + off, wrapped into [0,L),
// linear interp on x_proj (w1 zeroed when i0+1 == L, index clamped),
// mask-weighted sum over K taps.
// ---------------------------------------------------------------------------
__global__ __launch_bounds__(256) void dcn_sample_kernel(
    const float* __restrict__ xp, const float* __restrict__ off,
    const float* __restrict__ msk, float* __restrict__ out)
{
  const size_t idx = (size_t)blockIdx.x * blockDim.x + threadIdx.x;
  const int    c  = (int)(idx & (C_DIM - 1));
  const size_t nl = idx >> 8;
  const int    l  = (int)(nl & (L_DIM - 1));
  const size_t n  = nl >> 12;
  const int    g  = c >> 5;                       // GC = 32

  const float* op = off + nl * (G_DIM * K_TAPS) + g * K_TAPS;
  const float* mp = msk + nl * (G_DIM * K_TAPS) + g * K_TAPS;
  const float* xb = xp + ((n << 12) * C_DIM) + c; // &x_proj[n,0,c]

  float acc = 0.f;
#pragma unroll
  for (int k = 0; k < K_TAPS; ++k) {
    float loc = (float)l + (float)(k - 1) + op[k];
    float r = fmodf(loc, (float)L_DIM);
    if (r < 0.f)             r += (float)L_DIM;
    if (r >= (float)L_DIM)   r -= (float)L_DIM;   // fp rounding guard
    const float f0 = floorf(r);
    int   i0 = (int)f0;
    float w1 = r - f0;
    int   i1 = i0 + 1;
    if (i1 >= L_DIM) { w1 = 0.f; i1 = L_DIM - 1; }
    const float v = (1.f - w1) * xb[(size_t)i0 * C_DIM]
                  +        w1  * xb[(size_t)i1 * C_DIM];
    acc += mp[k] * v;
  }
  out[idx] = acc;
}

// ---------------------------------------------------------------------------
extern "C" void kernel_launch(void* const* d_in, const int* in_sizes, int n_in,
                              void* d_out, int out_size, void* d_ws, size_t ws_size,
                              hipStream_t stream) {
  const float* x     = (const float*)d_in[0];
  const float* W_in  = (const float*)d_in[1];
  const float* b_in  = (const float*)d_in[2];
  const float* W_out = (const float*)d_in[3];
  const float* b_out = (const float*)d_in[4];
  const float* W_red = (const float*)d_in[5];
  const float* b_red = (const float*)d_in[6];
  const float* W_off = (const float*)d_in[7];
  const float* b_off = (const float*)d_in[8];
  const float* W_msk = (const float*)d_in[9];
  const float* b_msk = (const float*)d_in[10];
  const float* dw_w  = (const float*)d_in[11];
  const float* dw_b  = (const float*)d_in[12];

  // workspace layout (fp32): xp | feat | red | off | msk ; samp reuses feat
  float* xp   = (float*)d_ws;
  float* feat = xp   + (size_t)M_DIM * C_DIM;
  float* red  = feat + (size_t)M_DIM * C_DIM;
  float* offb = red  + (size_t)M_DIM * (C_DIM / 2);
  float* mskb = offb + (size_t)M_DIM * (G_DIM * K_TAPS);
  float* samp = feat;   // feat is dead after the 'red' GEMM

  const int elem_blocks = (M_DIM * C_DIM) / 256;
  auto gemm_blocks = [](int ntilesN) { return ((M_DIM / 16) * ntilesN + 7) / 8; };

  // 1) x_proj = x @ W_in.T + b_in                  [65536 x 256], K=256
  gemm_nt_wmma_f32<256, 256><<<gemm_blocks(4), 256, 0, stream>>>(x, W_in, b_in, xp);
  // 2) depthwise conv -> feat
  dwconv3_kernel<<<elem_blocks, 256, 0, stream>>>(x, dw_w, dw_b, feat);
  // 3) red = feat @ W_red.T + b_red                [65536 x 128], K=256
  gemm_nt_wmma_f32<128, 256><<<gemm_blocks(2), 256, 0, stream>>>(feat, W_red, b_red, red);
  // 4) offset = red @ W_off.T + b_off              [65536 x 24],  K=128
  gemm_nt_wmma_f32<24, 128><<<gemm_blocks(1), 256, 0, stream>>>(red, W_off, b_off, offb);
  // 5) mask = red @ W_msk.T + b_msk                [65536 x 24],  K=128
  gemm_nt_wmma_f32<24, 128><<<gemm_blocks(1), 256, 0, stream>>>(red, W_msk, b_msk, mskb);
  // 6) deformable sampling -> samp
  dcn_sample_kernel<<<elem_blocks, 256, 0, stream>>>(xp, offb, mskb, samp);
  // 7) out = samp @ W_out.T + b_out                [65536 x 256], K=256
  gemm_nt_wmma_f32<256, 256><<<gemm_blocks(4), 256, 0, stream>>>(samp, W_out, b_out, (float*)d_out);
}